// ADDLoss_3676492005777
// MI455X (gfx1250) — compile-verified
//
#include <hip/hip_runtime.h>
#include <hip/hip_bf16.h>
#include <math.h>

#define N_OBJECTS 30
#define N_POINTS  500
#define BATCH     16384

typedef float v2f __attribute__((ext_vector_type(2)));
typedef float v8f __attribute__((ext_vector_type(8)));

// One wave (32 threads) per batch element.
// WMMA F32 16x16x4:  A = dR (rows 0..2 used, K padded to 4), B = 16 points (coords on K),
// D rows 0..2 give rot_diff x/y/z of point n in lane n (lanes 0..15).
__global__ void __launch_bounds__(32)
add_loss_wmma_kernel(const float* __restrict__ pred_r,
                     const float* __restrict__ pred_t,
                     const float* __restrict__ gt_r,
                     const float* __restrict__ gt_t,
                     const int*   __restrict__ obj_ids,
                     const float* __restrict__ points,
                     float*       __restrict__ per_batch)
{
    const int b    = blockIdx.x;
    const int lane = threadIdx.x;        // 0..31 (wave32)
    const int m    = lane & 15;          // row / point-in-tile
    const int half = lane >> 4;          // 0: K={0,1}, 1: K={2,3}

    // ---- quaternion -> rotation matrices (raw quats, matching reference) ----
    const float pw = pred_r[4*b+0], px = pred_r[4*b+1], py = pred_r[4*b+2], pz = pred_r[4*b+3];
    const float gw = gt_r[4*b+0],   gx = gt_r[4*b+1],   gy = gt_r[4*b+2],   gz = gt_r[4*b+3];

    const float p00 = 1.f - 2.f*(py*py + pz*pz), p01 = 2.f*(px*py - pw*pz), p02 = 2.f*(px*pz + pw*py);
    const float p10 = 2.f*(px*py + pw*pz), p11 = 1.f - 2.f*(px*px + pz*pz), p12 = 2.f*(py*pz - pw*px);
    const float p20 = 2.f*(px*pz - pw*py), p21 = 2.f*(py*pz + pw*px), p22 = 1.f - 2.f*(px*px + py*py);

    const float g00 = 1.f - 2.f*(gy*gy + gz*gz), g01 = 2.f*(gx*gy - gw*gz), g02 = 2.f*(gx*gz + gw*gy);
    const float g10 = 2.f*(gx*gy + gw*gz), g11 = 1.f - 2.f*(gx*gx + gz*gz), g12 = 2.f*(gy*gz - gw*gx);
    const float g20 = 2.f*(gx*gz - gw*gy), g21 = 2.f*(gy*gz + gw*gx), g22 = 1.f - 2.f*(gx*gx + gy*gy);

    const float d00 = p00 - g00, d01 = p01 - g01, d02 = p02 - g02;
    const float d10 = p10 - g10, d11 = p11 - g11, d12 = p12 - g12;
    const float d20 = p20 - g20, d21 = p21 - g21, d22 = p22 - g22;

    // ---- A operand: 16x4 f32. Lane m = row m; half 0 -> K{0,1}, half 1 -> K{2,3}(K3=0) ----
    float a0 = 0.f, a1 = 0.f;
    if (half == 0) {
        if      (m == 0) { a0 = d00; a1 = d01; }
        else if (m == 1) { a0 = d10; a1 = d11; }
        else if (m == 2) { a0 = d20; a1 = d21; }
    } else {
        if      (m == 0) { a0 = d02; }
        else if (m == 1) { a0 = d12; }
        else if (m == 2) { a0 = d22; }
    }
    v2f A; A.x = a0; A.y = a1;

    // ---- translation diff (uniform per block) ----
    const float tdx = pred_t[3*b+0] - gt_t[3*b+0];
    const float tdy = pred_t[3*b+1] - gt_t[3*b+1];
    const float tdz = pred_t[3*b+2] - gt_t[3*b+2];
    const float trans = __builtin_amdgcn_sqrtf(tdx*tdx + tdy*tdy + tdz*tdz);

    const float* __restrict__ pts = points + (size_t)obj_ids[b] * (N_POINTS * 3);
    __builtin_prefetch(pts, 0, 1);                 // global_prefetch of the 6KB point block
    __builtin_prefetch(pts + 1024, 0, 1);

    float add_sum = 0.f, rot_sum = 0.f;

    // 500 points -> 32 tiles of 16 (tail masked by value, not by EXEC)
    #pragma unroll 4
    for (int tile = 0; tile < (N_POINTS + 15) / 16; ++tile) {
        const int  p    = tile * 16 + m;
        const bool inb  = (p < N_POINTS);

        // Branchless, always-in-bounds loads:
        //   half 0 wants {x,y} = pts[3p], pts[3p+1];  half 1 wants {z,0} = pts[3p+2], 0.
        const int idx0 = inb ? (3*p + 2*half) : 0;     // <= 1499, always valid
        const int idx1 = (inb && half == 0) ? (3*p + 1) : 0;
        const float v0 = pts[idx0];
        const float v1 = pts[idx1];

        v2f B;
        B.x = inb ? v0 : 0.f;
        B.y = (inb && half == 0) ? v1 : 0.f;

        v8f C = {};
        C = __builtin_amdgcn_wmma_f32_16x16x4_f32(
                /*neg_a=*/false, A, /*neg_b=*/false, B,
                /*c_mod=*/(short)0, C, /*reuse_a=*/false, /*reuse_b=*/false);

        // D: lane n<16, VGPR r holds D[r][n] -> rot_diff components of point n.
        const float mask = (inb && half == 0) ? 1.f : 0.f;
        const float rx = C[0], ry = C[1], rz = C[2];
        rot_sum = __builtin_fmaf(mask,
                                 __builtin_amdgcn_sqrtf(rx*rx + ry*ry + rz*rz),
                                 rot_sum);
        const float ax = rx + tdx, ay = ry + tdy, az = rz + tdz;
        add_sum = __builtin_fmaf(mask,
                                 __builtin_amdgcn_sqrtf(ax*ax + ay*ay + az*az),
                                 add_sum);
    }

    // wave reduction: nonzero data lives only in lanes 0..15
    #pragma unroll
    for (int off = 8; off > 0; off >>= 1) {
        add_sum += __shfl_down(add_sum, off, 16);
        rot_sum += __shfl_down(rot_sum, off, 16);
    }

    if (lane == 0) {
        per_batch[b] = (add_sum + rot_sum) * (1.f / N_POINTS) + trans;
    }
}

// Deterministic fixed-order tree reduction of the 16384 per-batch losses.
__global__ void __launch_bounds__(256)
final_reduce_kernel(const float* __restrict__ per_batch, float* __restrict__ out)
{
    __shared__ float smem[256];
    float acc = 0.f;
    for (int i = threadIdx.x; i < BATCH; i += 256)
        acc += per_batch[i];
    smem[threadIdx.x] = acc;
    __syncthreads();
    #pragma unroll
    for (int off = 128; off > 0; off >>= 1) {
        if (threadIdx.x < off) smem[threadIdx.x] += smem[threadIdx.x + off];
        __syncthreads();
    }
    if (threadIdx.x == 0)
        out[0] = smem[0] * (1.f / BATCH);
}

extern "C" void kernel_launch(void* const* d_in, const int* in_sizes, int n_in,
                              void* d_out, int out_size, void* d_ws, size_t ws_size,
                              hipStream_t stream)
{
    const float* pred_r  = (const float*)d_in[0];
    const float* pred_t  = (const float*)d_in[1];
    const float* gt_r    = (const float*)d_in[2];
    const float* gt_t    = (const float*)d_in[3];
    const int*   obj_ids = (const int*)d_in[4];
    const float* points  = (const float*)d_in[5];

    float* per_batch = (float*)d_ws;          // BATCH floats = 64 KB scratch
    float* out       = (float*)d_out;

    add_loss_wmma_kernel<<<BATCH, 32, 0, stream>>>(pred_r, pred_t, gt_r, gt_t,
                                                   obj_ids, points, per_batch);
    final_reduce_kernel<<<1, 256, 0, stream>>>(per_batch, out);
}